// HyperbolicMemory_5411658793259
// MI455X (gfx1250) — compile-verified
//
#include <hip/hip_runtime.h>
#include <hip/hip_bf16.h>
#include <math.h>

typedef __attribute__((ext_vector_type(2))) float v2f;
typedef __attribute__((ext_vector_type(4))) float v4f;
typedef __attribute__((ext_vector_type(8))) float v8f;

#define D_DIM 128
#define TOPK 16
#define EPS_F 1e-6f

// ---------------------------------------------------------------------------
// Kernel 1: squared norms of query rows (B) and store rows (N)
// ---------------------------------------------------------------------------
__global__ void norms_kernel(const float* __restrict__ q,
                             const float* __restrict__ y,
                             float* __restrict__ x2,
                             float* __restrict__ y2,
                             int B, int N) {
    int i = blockIdx.x * blockDim.x + threadIdx.x;
    if (i < B) {
        const v4f* row = (const v4f*)(q + (size_t)i * D_DIM);
        float s = 0.0f;
        #pragma unroll
        for (int j = 0; j < D_DIM / 4; ++j) {
            v4f v = row[j];
            s += v.x * v.x + v.y * v.y + v.z * v.z + v.w * v.w;
        }
        x2[i] = s;
    } else if (i < B + N) {
        int r = i - B;
        const v4f* row = (const v4f*)(y + (size_t)r * D_DIM);
        float s = 0.0f;
        #pragma unroll
        for (int j = 0; j < D_DIM / 4; ++j) {
            v4f v = row[j];
            s += v.x * v.x + v.y * v.y + v.z * v.z + v.w * v.w;
        }
        y2[r] = s;
    }
}

// ---------------------------------------------------------------------------
// Kernel 2: Poincare distance matrix via V_WMMA_F32_16X16X4_F32.
// One wave32 per 16x16 output tile; 8 waves (256 threads) per block.
// A (16x4 f32): lane<16 -> M=lane, K={kk,kk+1}; lane>=16 -> M=lane-16, K={kk+2,kk+3}
// B (4x16 f32): mirrored (N=lane&15, K pair by lane half) -> load y rows directly (B = Y^T)
// C/D (16x16 f32): VGPR v: M = v + 8*(lane>=16), N = lane&15
// ---------------------------------------------------------------------------
__global__ void dist_wmma_kernel(const float* __restrict__ q,    // (B,128)
                                 const float* __restrict__ y,    // (N,128)
                                 const float* __restrict__ x2,   // (B)
                                 const float* __restrict__ y2,   // (N)
                                 float* __restrict__ dists,      // (B,N)
                                 int N) {
    const int lane = threadIdx.x & 31;
    const int wave = threadIdx.x >> 5;
    const long long tile = (long long)blockIdx.x * 8 + wave;
    const int tiles_n = N / 16;
    const int tn = (int)(tile % tiles_n);
    const int tm = (int)(tile / tiles_n);
    const int m0 = tm * 16;
    const int n0 = tn * 16;

    const int half = lane >> 4;   // which K sub-pair this lane holds
    const int l15  = lane & 15;

    const float* qrow = q + (size_t)(m0 + l15) * D_DIM;  // A row M=l15
    const float* yrow = y + (size_t)(n0 + l15) * D_DIM;  // B col N=l15 (y row-major)

    v8f c = {};
    #pragma unroll
    for (int kk = 0; kk < D_DIM; kk += 4) {
        const int kb = kk + half * 2;
        v2f a = *(const v2f*)(qrow + kb);
        v2f b = *(const v2f*)(yrow + kb);
        // D = A x B + C : 8-arg form (neg_a, A, neg_b, B, c_mod, C, reuse_a, reuse_b)
        c = __builtin_amdgcn_wmma_f32_16x16x4_f32(false, a, false, b,
                                                  (short)0, c, false, false);
    }

    const int n = n0 + l15;
    const float yn2 = y2[n];
    #pragma unroll
    for (int v = 0; v < 8; ++v) {
        const int m = m0 + v + (half ? 8 : 0);
        const float xm2  = x2[m];
        const float dot  = c[v];
        float diff2 = fmaxf(xm2 + yn2 - 2.0f * dot, 0.0f);
        float den   = fmaxf((1.0f - xm2) * (1.0f - yn2), EPS_F);
        float arg   = fmaxf(1.0f + 2.0f * diff2 / den, 1.0f + EPS_F);
        // arccosh(arg)
        float dist  = logf(arg + sqrtf(arg * arg - 1.0f));
        dists[(size_t)m * N + n] = dist;
    }
}

// ---------------------------------------------------------------------------
// Kernel 3: per-query top-16 (smallest distance, ties -> lowest index).
// 256 threads/block; register insertion lists, LDS merge by thread 0.
// ---------------------------------------------------------------------------
#define TPB_TOPK 256
__global__ void topk_kernel(const float* __restrict__ dists, int N,
                            float* __restrict__ out_vals,   // (B,16) into d_out
                            float* __restrict__ out_idxf,   // (B,16) into d_out
                            int* __restrict__ tki) {        // (B,16) ws (int)
    const int b = blockIdx.x;
    const float* row = dists + (size_t)b * N;

    float v[TOPK];
    int   id[TOPK];
    #pragma unroll
    for (int i = 0; i < TOPK; ++i) { v[i] = 3.402823e38f; id[i] = 0x7fffffff; }

    for (int j = threadIdx.x; j < N; j += blockDim.x) {
        float d = row[j];
        if (d < v[TOPK - 1]) {
            #pragma unroll
            for (int i = TOPK - 1; i >= 0; --i) {
                if (i > 0 && (v[i - 1] > d)) {
                    v[i] = v[i - 1]; id[i] = id[i - 1];
                } else {
                    v[i] = d; id[i] = j;
                    break;
                }
            }
        }
    }

    __shared__ float sv[TPB_TOPK * TOPK];
    __shared__ int   si[TPB_TOPK * TOPK];
    #pragma unroll
    for (int i = 0; i < TOPK; ++i) {
        sv[threadIdx.x * TOPK + i] = v[i];
        si[threadIdx.x * TOPK + i] = id[i];
    }
    __syncthreads();

    if (threadIdx.x == 0) {
        float bv[TOPK]; int bi[TOPK];
        #pragma unroll
        for (int i = 0; i < TOPK; ++i) { bv[i] = 3.402823e38f; bi[i] = 0x7fffffff; }
        for (int t = 0; t < TPB_TOPK * TOPK; ++t) {
            float d = sv[t];
            int   j = si[t];
            bool better_last = (d < bv[TOPK - 1]) ||
                               (d == bv[TOPK - 1] && j < bi[TOPK - 1]);
            if (better_last) {
                #pragma unroll
                for (int i = TOPK - 1; i >= 0; --i) {
                    bool shift = (i > 0) &&
                                 ((bv[i - 1] > d) || (bv[i - 1] == d && bi[i - 1] > j));
                    if (shift) {
                        bv[i] = bv[i - 1]; bi[i] = bi[i - 1];
                    } else {
                        bv[i] = d; bi[i] = j;
                        break;
                    }
                }
            }
        }
        #pragma unroll
        for (int i = 0; i < TOPK; ++i) {
            out_vals[(size_t)b * TOPK + i] = bv[i];
            out_idxf[(size_t)b * TOPK + i] = (float)bi[i];
            tki[(size_t)b * TOPK + i]      = bi[i];
        }
    }
}

// ---------------------------------------------------------------------------
// Kernel 4: gather q_store / m_store / p_store rows. One block per (b,k) pair.
// ---------------------------------------------------------------------------
__global__ void gather_small_kernel(const float* __restrict__ qs,
                                    const float* __restrict__ ms,
                                    const float* __restrict__ ps,
                                    const int* __restrict__ tki,
                                    float* __restrict__ out_q,
                                    float* __restrict__ out_m,
                                    float* __restrict__ out_p) {
    const int pair = blockIdx.x;       // b*16 + j
    const int idx  = tki[pair];
    const int t    = threadIdx.x;      // 0..127
    out_q[(size_t)pair * D_DIM + t] = qs[(size_t)idx * D_DIM + t];
    out_p[(size_t)pair * D_DIM + t] = ps[(size_t)idx * D_DIM + t];
    if (t == 0) out_m[pair] = ms[idx];
}

// ---------------------------------------------------------------------------
// Kernel 5: gather J rows: 8192 x 64KB streaming copies. Non-temporal 128-bit
// ops (clang ext-vector so the builtin accepts it): ~1 GB of traffic that must
// not pollute L2.
// ---------------------------------------------------------------------------
__global__ void gather_J_kernel(const float* __restrict__ Js,
                                const int* __restrict__ tki,
                                float* __restrict__ out_J) {
    const int pair = blockIdx.x;
    const int idx  = tki[pair];
    const v4f* __restrict__ src = (const v4f*)(Js + (size_t)idx * (D_DIM * D_DIM));
    v4f* __restrict__ dst       = (v4f*)(out_J + (size_t)pair * (D_DIM * D_DIM));
    const int nvec = (D_DIM * D_DIM) / 4;   // 4096
    for (int i = threadIdx.x; i < nvec; i += blockDim.x) {
        v4f v = __builtin_nontemporal_load(src + i);
        __builtin_nontemporal_store(v, dst + i);
    }
}

// ---------------------------------------------------------------------------
// Host launcher
// ---------------------------------------------------------------------------
extern "C" void kernel_launch(void* const* d_in, const int* in_sizes, int n_in,
                              void* d_out, int out_size, void* d_ws, size_t ws_size,
                              hipStream_t stream) {
    const float* q_query = (const float*)d_in[0];   // (B,128)
    const float* q_store = (const float*)d_in[1];   // (N,128)
    const float* m_store = (const float*)d_in[2];   // (N,1)
    const float* J_store = (const float*)d_in[3];   // (N,128,128)
    const float* p_store = (const float*)d_in[4];   // (N,128)
    // d_in[5] is k (==16), fixed at compile time as TOPK.

    const int B = in_sizes[0] / D_DIM;              // 512
    const int N = in_sizes[1] / D_DIM;              // 16384

    // Workspace layout (bytes)
    char* ws = (char*)d_ws;
    float* dists = (float*)ws;                                  // B*N f32
    float* x2    = (float*)(ws + (size_t)B * N * 4);            // B f32
    float* y2    = x2 + B;                                      // N f32
    int*   tki   = (int*)(y2 + N);                              // B*16 i32

    // Output layout (flat concat, reference return order)
    float* out = (float*)d_out;
    long long BK    = (long long)B * TOPK;
    float* out_q    = out;                                       // B*16*128
    float* out_m    = out_q + BK * D_DIM;                        // B*16
    float* out_J    = out_m + BK;                                // B*16*128*128
    float* out_p    = out_J + BK * (long long)(D_DIM * D_DIM);   // B*16*128
    float* out_vals = out_p + BK * D_DIM;                        // B*16
    float* out_idx  = out_vals + BK;                             // B*16

    // 1) norms
    {
        int total = B + N;
        int blocks = (total + 255) / 256;
        norms_kernel<<<blocks, 256, 0, stream>>>(q_query, q_store, x2, y2, B, N);
    }
    // 2) distance matrix via f32 WMMA (one wave per 16x16 tile, 8 waves/block)
    {
        long long tiles = (long long)(B / 16) * (N / 16);        // 32768
        int blocks = (int)(tiles / 8);                           // 4096
        dist_wmma_kernel<<<blocks, 256, 0, stream>>>(q_query, q_store, x2, y2, dists, N);
    }
    // 3) top-16 per query
    topk_kernel<<<B, TPB_TOPK, 0, stream>>>(dists, N, out_vals, out_idx, tki);
    // 4) small gathers
    gather_small_kernel<<<(int)BK, D_DIM, 0, stream>>>(q_store, m_store, p_store,
                                                       tki, out_q, out_m, out_p);
    // 5) J gather (bandwidth-bound: ~1 GB @ 23.3 TB/s ~= 46 us floor)
    gather_J_kernel<<<(int)BK, 256, 0, stream>>>(J_store, tki, out_J);
}